// SAGE_39822936768930
// MI455X (gfx1250) — compile-verified
//
#include <hip/hip_runtime.h>

// ---------------------------------------------------------------------------
// GraphSAGE 2-layer inference for MI455X (gfx1250, wave32).
//   layer: out = (segment_sum(x[src],dst)/max(deg,1)) @ w_l + x @ w_r + b
//
// Phase structure:
//   * aggregation: one wave per edge, float4 per lane; gathers are coalesced
//     512B bursts, scatter-adds are contiguous f32 atomics serviced in L2
//     (entire working set ~150MB fits the 192MB L2 -> L2-atomic bound).
//   * GEMMs: fp32 WMMA (v_wmma_f32_16x16x4_f32) — total compute is ~9 GFLOP,
//     nowhere near the matrix ceiling, so we keep full reference precision.
//     k-outer / n-tile-inner with NT live accumulators; weights staged in LDS
//     (pair-interleaved so each B operand is a single ds_load_b64), shared by
//     all 8 waves of the block -> 8x less global B/A traffic than naive.
// ---------------------------------------------------------------------------

typedef __attribute__((ext_vector_type(2))) float v2f;
typedef __attribute__((ext_vector_type(8))) float v8f;

__device__ __forceinline__ v8f wmma_f32(v2f a, v2f b, v8f c) {
  // 8 args: (neg_a, A, neg_b, B, c_mod, C, reuse_a, reuse_b)
  return __builtin_amdgcn_wmma_f32_16x16x4_f32(false, a, false, b,
                                               (short)0, c, false, false);
}

// ------------------------------ utility ------------------------------------
__global__ void k_zero_f32(float* __restrict__ p, long n) {
  long i = (long)blockIdx.x * blockDim.x + threadIdx.x;
  if (i < n) p[i] = 0.0f;
}

// one thread per edge: deg[dst] += 1
__global__ void k_degree(const int* __restrict__ dst, float* __restrict__ deg, int E) {
  int e = blockIdx.x * blockDim.x + threadIdx.x;
  if (e < E) atomicAdd(&deg[dst[e]], 1.0f);
}

// in place: deg -> 1/max(deg,1)
__global__ void k_invdeg(float* __restrict__ deg, int n) {
  int i = blockIdx.x * blockDim.x + threadIdx.x;
  if (i < n) deg[i] = 1.0f / fmaxf(deg[i], 1.0f);
}

// one wave (32 lanes) per edge; lane handles 4 contiguous feature floats.
__global__ void k_scatter_add(const float* __restrict__ feat,
                              const int* __restrict__ src,
                              const int* __restrict__ dst,
                              float* __restrict__ agg, int E) {
  long t = (long)blockIdx.x * blockDim.x + threadIdx.x;
  long e = t >> 5;
  int lane = (int)(t & 31);
  if (e >= E) return;
  int s = src[e];
  int d = dst[e];
  int c = lane * 4;
  const float4 v = *(const float4*)(feat + (long)s * 128 + c);
  float* a = agg + (long)d * 128 + c;
  atomicAdd(a + 0, v.x);
  atomicAdd(a + 1, v.y);
  atomicAdd(a + 2, v.z);
  atomicAdd(a + 3, v.w);
}

// ------------------------- fused SAGE GEMM ---------------------------------
// One wave owns a 16-row tile of the output:
//   out[m0:m0+16, :] = relu?((agg*inv) @ wl + xin @ wr + bias)
// K=128 processed in two 64-row phases; each phase stages 32 pair-rows of
// wl and wr in LDS (pair-interleaved: lds[p*NPAD*2 + n*2 + (k&1)]), then the
// wave streams A once and issues 2*NT chained v_wmma_f32_16x16x4_f32 per
// 4-wide K step. NOUT=128 (layer1, RELU) or 41 (layer2, padded to 48).
template <int NOUT, bool RELU>
__global__ void k_sage_gemm(const float* __restrict__ agg,
                            const float* __restrict__ invdeg,
                            const float* __restrict__ xin,
                            const float* __restrict__ wl,
                            const float* __restrict__ wr,
                            const float* __restrict__ bias,
                            float* __restrict__ out, int nrows) {
  constexpr int K    = 128;
  constexpr int NT   = (NOUT + 15) / 16;
  constexpr int NPAD = NT * 16;
  constexpr int KH   = 64;        // K rows staged per phase
  constexpr int PPH  = KH / 2;    // pair-rows per phase (32)

  extern __shared__ float lds[];
  float* swl = lds;                           // PPH*NPAD*2 floats
  float* swr = lds + (long)PPH * NPAD * 2;    // PPH*NPAD*2 floats

  const int tid  = threadIdx.x;
  const int lane = tid & 31;
  const int wave = (int)((blockIdx.x * blockDim.x + tid) >> 5);
  const int m0   = wave * 16;
  const bool active = (m0 < nrows);           // wave-uniform

  const int half = lane >> 4;                 // K-pair selector (0/1)
  const int l16  = lane & 15;
  const int arow = m0 + l16;                  // A-operand row for this lane
  const int ak0  = half * 2;

  float aiv = 0.0f;
  const float* aP = agg;
  const float* xP = xin;
  if (active) {
    aiv = invdeg[arow];
    aP  = agg + (long)arow * K;
    xP  = xin + (long)arow * K;
  }

  v8f c[NT] = {};                             // NT live 16x16 accumulators

  for (int kh = 0; kh < K; kh += KH) {
    __syncthreads();                          // previous phase fully consumed
    // cooperative stage of weight rows [kh, kh+KH), pair-interleaved + padded
    for (int i = tid; i < PPH * NPAD; i += 256) {
      const int p = i / NPAD;
      const int n = i - p * NPAD;
      const int krow = kh + 2 * p;
      float2 vl = make_float2(0.0f, 0.0f);
      float2 vr = vl;
      if (n < NOUT) {
        vl.x = wl[(long)krow * NOUT + n];
        vl.y = wl[(long)(krow + 1) * NOUT + n];
        vr.x = wr[(long)krow * NOUT + n];
        vr.y = wr[(long)(krow + 1) * NOUT + n];
      }
      *(float2*)(swl + (long)p * NPAD * 2 + n * 2) = vl;
      *(float2*)(swr + (long)p * NPAD * 2 + n * 2) = vr;
    }
    __syncthreads();

    if (active) {                             // wave-uniform: EXEC all-ones
#pragma unroll 2
      for (int k = 0; k < KH; k += 4) {
        const float2 a2 = *(const float2*)(aP + kh + k + ak0);
        const float2 x2 = *(const float2*)(xP + kh + k + ak0);
        v2f am = {a2.x * aiv, a2.y * aiv};    // mean-neighbor path
        v2f ax = {x2.x, x2.y};                // self path
        const int p = (k >> 1) + half;
        const float* blP = swl + (long)p * NPAD * 2 + l16 * 2;
        const float* brP = swr + (long)p * NPAD * 2 + l16 * 2;
#pragma unroll
        for (int nt = 0; nt < NT; ++nt) {
          const float2 bl2 = *(const float2*)(blP + nt * 32);
          const float2 br2 = *(const float2*)(brP + nt * 32);
          v2f bl = {bl2.x, bl2.y};
          v2f br = {br2.x, br2.y};
          c[nt] = wmma_f32(am, bl, c[nt]);
          c[nt] = wmma_f32(ax, br, c[nt]);
        }
      }
    }
  }

  if (!active) return;
  // C layout: VGPR j -> row m0 + 8*half + j, column nt*16 + l16
#pragma unroll
  for (int nt = 0; nt < NT; ++nt) {
    const int n = nt * 16 + l16;
    if (n < NOUT) {
      const float bv = bias[n];
#pragma unroll
      for (int j = 0; j < 8; ++j) {
        const int row = m0 + half * 8 + j;
        float v = c[nt][j] + bv;
        if (RELU) v = fmaxf(v, 0.0f);
        out[(long)row * NOUT + n] = v;
      }
    }
  }
}

// ------------------------------ launcher -----------------------------------
extern "C" void kernel_launch(void* const* d_in, const int* in_sizes, int n_in,
                              void* d_out, int out_size, void* d_ws, size_t ws_size,
                              hipStream_t stream) {
  const float* x   = (const float*)d_in[0];
  const int*   ei  = (const int*)d_in[1];
  const float* w1l = (const float*)d_in[2];
  const float* w1r = (const float*)d_in[3];
  const float* b1  = (const float*)d_in[4];
  const float* w2l = (const float*)d_in[5];
  const float* w2r = (const float*)d_in[6];
  const float* b2  = (const float*)d_in[7];
  float* out = (float*)d_out;

  const int N = in_sizes[0] / 128;   // 100000
  const int E = in_sizes[1] / 2;     // 1600000
  const int* src = ei;
  const int* dst = ei + E;

  // workspace layout: deg/inv | agg (reused both layers) | h
  char* ws = (char*)d_ws;
  float* deg = (float*)ws;
  size_t off1 = ((size_t)N * 4 + 255) & ~(size_t)255;
  float* agg = (float*)(ws + off1);
  size_t off2 = off1 + ((((size_t)N * 128 * 4) + 255) & ~(size_t)255);
  float* h = (float*)(ws + off2);

  const long nAgg = (long)N * 128;
  const int zb   = (int)((nAgg + 255) / 256);
  const int eb   = (E + 255) / 256;
  const int nb   = (N + 255) / 256;
  const long sct = (long)E * 32;
  const int sb   = (int)((sct + 255) / 256);
  const int waves = (N + 15) / 16;               // 16 rows per wave
  const int gb    = (waves * 32 + 255) / 256;    // 8 waves / 256-thread block

  // dynamic LDS: one phase's worth of both weight matrices, pair-interleaved
  const size_t sh1 = (size_t)2 * 32 * 128 * 2 * sizeof(float); // 65536 B
  const size_t sh2 = (size_t)2 * 32 * 48  * 2 * sizeof(float); // 24576 B

  // ---- layer 1 ----
  k_zero_f32<<<nb, 256, 0, stream>>>(deg, N);
  k_zero_f32<<<zb, 256, 0, stream>>>(agg, nAgg);
  k_degree<<<eb, 256, 0, stream>>>(dst, deg, E);
  k_invdeg<<<nb, 256, 0, stream>>>(deg, N);
  k_scatter_add<<<sb, 256, 0, stream>>>(x, src, dst, agg, E);
  k_sage_gemm<128, true><<<gb, 256, sh1, stream>>>(agg, deg, x, w1l, w1r, b1, h, N);

  // ---- layer 2 ----
  k_zero_f32<<<zb, 256, 0, stream>>>(agg, nAgg);
  k_scatter_add<<<sb, 256, 0, stream>>>(h, src, dst, agg, E);
  k_sage_gemm<41, false><<<gb, 256, sh2, stream>>>(agg, deg, h, w2l, w2r, b2, out, N);
}